// ISTFTModule_47906065219848
// MI455X (gfx1250) — compile-verified
//
#include <hip/hip_runtime.h>
#include <hip/hip_bf16.h>

// ---------------- problem constants (from the reference) ----------------
#define N_FFT   1024
#define HOP     256
#define NFREQ   513            // N_FFT/2 + 1
#define KDIM    1026           // 2*NFREQ
#define KPAD    1028           // pad K to multiple of 4 (zero rows)
#define BATCH   16
#define TFRAMES 1024
#define LOUT    ((TFRAMES - 1) * HOP + N_FFT)   // 262912

#define LDS_A_FLOATS (KPAD * 16)                // 16448 floats = 65792 B

typedef __attribute__((ext_vector_type(2))) float v2f;
typedef __attribute__((ext_vector_type(8))) float v8f;

// ---------------------------------------------------------------------------
// Kernel 1: build the padded, scaled windowed-DFT kernel matrix in workspace,
// stored PAIR-INTERLEAVED on K:  kerp[c/2][n][2]  (so a (K,K+1) B-fragment
// pair is one 8-byte load).
//   real rows (c<513):     cos(2*pi*c*n/N) * win[n] * s_c * coeff / N
//   imag rows (513..1025): -sin(2*pi*f*n/N) * win[n] * 2 * coeff / N
//   rows 1026,1027: zero padding
//   coeff = HOP / sum(win^2) = 256/384 = 2/3 folded in (no post-scale pass).
// ---------------------------------------------------------------------------
__global__ void __launch_bounds__(256) istft_kergen(float* __restrict__ kerp) {
    const int idx = blockIdx.x * blockDim.x + threadIdx.x;
    if (idx >= KPAD * N_FFT) return;
    const int c = idx >> 10;          // / N_FFT
    const int n = idx & (N_FFT - 1);  // % N_FFT

    const float TWO_PI = 6.28318530717958647692f;
    const float coeff  = (float)HOP / (0.375f * (float)N_FFT);   // 2/3
    const float inv_n  = 1.0f / (float)N_FFT;

    // periodic Hann window
    const float win = 0.5f * (1.0f - __cosf(TWO_PI * (float)n * inv_n));

    float v = 0.0f;
    if (c < NFREQ) {
        const int f  = c;
        const int ph = (f * n) & (N_FFT - 1);    // exact mod-1024 phase
        const float s = (f >= 1 && f <= (N_FFT / 2 - 1)) ? 2.0f : 1.0f;
        v = __cosf(TWO_PI * (float)ph * inv_n) * win * s * coeff * inv_n;
    } else if (c < KDIM) {
        const int f  = c - NFREQ;
        const int ph = (f * n) & (N_FFT - 1);
        v = -__sinf(TWO_PI * (float)ph * inv_n) * win * 2.0f * coeff * inv_n;
    }
    // pair-interleaved store: [c/2][n][2]
    kerp[(((size_t)(c >> 1)) * N_FFT + n) * 2 + (c & 1)] = v;
}

// ---------------------------------------------------------------------------
// Kernel 2: zero the output (overlap-add accumulates into it).
// ---------------------------------------------------------------------------
__global__ void __launch_bounds__(256) istft_zero(float* __restrict__ out, int n) {
    int i = blockIdx.x * blockDim.x + threadIdx.x;
    if (i < n) out[i] = 0.0f;
}

// ---------------------------------------------------------------------------
// Kernel 3: fused GEMM (fp32 WMMA 16x16x4) + overlap-add epilogue.
//   grid:  BATCH * (TFRAMES/16) = 1024 workgroups
//   block: 256 threads = 8 wave32 waves; wave w owns columns [w*128, w*128+128)
//
//   Phase 0 (once per WG): stage the 16-row A tile into LDS, pair-interleaved
//   on K: alds[c/2][m][2]. The real/imag gather and K-pad happen HERE, not in
//   the k-loop. Staging loads are coalesced float2 reads of x[b][f][t][0..1].
//
//   k-loop (step 4): 1 ds_load (A pair) + 8 global_load_b64 (B pairs from the
//   L2-resident ker; no prefetch — 4.2 MB ker lives in the 192 MB L2) +
//   8 v_wmma_f32_16x16x4_f32.
//
//   Fragment layouts per CDNA5 ISA (CDNA5 ISA):
//     A (16x4 f32): lanes 0-15 -> K={k,k+1}, lanes 16-31 -> K={k+2,k+3};
//                   M = lane % 16.  B mirrors with N = lane % 16.
//     C/D (16x16 f32, 8 VGPRs): VGPR v, lane L -> M = v + (L>=16 ? 8 : 0),
//                               N = L % 16.
//
//   Epilogue: frames[b][t][n] -> wav[b][t*HOP + n]; 4-way frame overlap
//   resolved with f32 atomics at L2 (coeff already folded into ker).
// ---------------------------------------------------------------------------
__global__ void __launch_bounds__(256) istft_wmma_gemm(const float* __restrict__ x,
                                                       const float* __restrict__ kerp,
                                                       float* __restrict__ out) {
    extern __shared__ float alds[];              // LDS_A_FLOATS floats

    const int lane = threadIdx.x & 31;
    const int wave = threadIdx.x >> 5;           // 0..7
    const int b    = blockIdx.x >> 6;            // / (TFRAMES/16)
    const int m0   = (blockIdx.x & 63) << 4;     // M-tile base (frame t)
    const int n0   = wave << 7;                  // wave's N-stripe base
    const int ncol = lane & 15;                  // B/C column within tile
    const int mrow = lane & 15;                  // A row within tile
    const int pk   = lane >> 4;                  // 0: K pair (k,k+1); 1: (k+2,k+3)

    // ---- Phase 0: stage A tile (16 x KPAD) into LDS, pair-interleaved ----
    const float* xb = x + (size_t)b * NFREQ * TFRAMES * 2;
    for (int i = threadIdx.x; i < NFREQ * 16; i += 256) {
        const int f = i >> 4;
        const int m = i & 15;
        // coalesced: consecutive threads -> consecutive t
        const float2 ri = *(const float2*)(xb + ((size_t)f * TFRAMES + (m0 + m)) * 2);
        const int cr = f;                        // real row
        const int ci = f + NFREQ;                // imag row
        alds[(cr >> 1) * 32 + m * 2 + (cr & 1)] = ri.x;
        alds[(ci >> 1) * 32 + m * 2 + (ci & 1)] = ri.y;
    }
    if (threadIdx.x < 32) {                      // zero K-pad rows 1026,1027
        const int c = KDIM + (threadIdx.x >> 4);
        const int m = threadIdx.x & 15;
        alds[(c >> 1) * 32 + m * 2 + (c & 1)] = 0.0f;
    }
    __syncthreads();

    v8f acc[8];
#pragma unroll
    for (int i = 0; i < 8; ++i)
        acc[i] = (v8f){0.f, 0.f, 0.f, 0.f, 0.f, 0.f, 0.f, 0.f};

    for (int k = 0; k < KDIM; k += 4) {
        const int p = (k >> 1) + pk;             // K-pair index for this lane

        // ---- A fragment: one 8-byte LDS load ----
        const v2f a = *(const v2f*)&alds[p * 32 + mrow * 2];

        // ---- B fragments: 8 x 8-byte loads from L2-resident ker ----
        const float* kp = kerp + ((size_t)p * N_FFT + n0 + ncol) * 2;

#pragma unroll
        for (int i = 0; i < 8; ++i) {
            const v2f bf = *(const v2f*)(kp + i * 32);
            acc[i] = __builtin_amdgcn_wmma_f32_16x16x4_f32(
                /*neg_a=*/false, a,
                /*neg_b=*/false, bf,
                /*c_mod=*/(short)0, acc[i],
                /*reuse_a=*/false, /*reuse_b=*/false);
        }
    }

    // ---- fused overlap-add epilogue (coeff already folded into ker) ----
    float* ob = out + (size_t)b * LOUT;
    const int mhi = (lane >> 4) ? 8 : 0;
#pragma unroll
    for (int i = 0; i < 8; ++i) {
        float* obn = ob + (size_t)(m0 + mhi) * HOP + (n0 + i * 16 + ncol);
#pragma unroll
        for (int v = 0; v < 8; ++v) {
            atomicAdd(obn + (size_t)v * HOP, acc[i][v]);   // frame t = m0+mhi+v
        }
    }
}

// ---------------------------------------------------------------------------
extern "C" void kernel_launch(void* const* d_in, const int* in_sizes, int n_in,
                              void* d_out, int out_size, void* d_ws, size_t ws_size,
                              hipStream_t stream) {
    const float* x  = (const float*)d_in[0];
    float* out      = (float*)d_out;
    float* kerp     = (float*)d_ws;              // KPAD*N_FFT floats = 4.21 MB

    // 1) build padded, pair-interleaved, scaled DFT kernel in workspace
    {
        const int total = KPAD * N_FFT;
        istft_kergen<<<(total + 255) / 256, 256, 0, stream>>>(kerp);
    }
    // 2) zero output (overlap-add accumulates)
    {
        istft_zero<<<(out_size + 255) / 256, 256, 0, stream>>>(out, out_size);
    }
    // 3) fused WMMA GEMM + overlap-add (dynamic LDS: 65792 B A-tile)
    {
        dim3 grid(BATCH * (TFRAMES / 16));
        istft_wmma_gemm<<<grid, 256, LDS_A_FLOATS * sizeof(float), stream>>>(x, kerp, out);
    }
}